// PredictiveCodingModel_44504451121175
// MI455X (gfx1250) — compile-verified
//
#include <hip/hip_runtime.h>
#include <hip/hip_bf16.h>
#include <math.h>

// ---------------------------------------------------------------------------
// PredictiveCodingModel: affine-linear complex recurrence, parallelized as a
// chunked linear scan.  Phase 1 (t<30000): y' = A y + g z (A constant 4x4).
// Phase 2 (t>=30000): y' = mu .* y (diagonal complex).  Outputs y, b, a, ysum
// with y/ysum scaled by 1/|y[0,30000,0]|.
//
// Phase-1 replay is a WMMA f32 GEMM per 32-step chunk; K and U tiles are
// staged into LDS by the Tensor Data Mover (TDM), with TDM LDS padding
// giving a conflict-free stride-68 layout.
// ---------------------------------------------------------------------------

typedef __attribute__((ext_vector_type(2))) float v2f;
typedef __attribute__((ext_vector_type(8))) float v8f;
typedef __attribute__((ext_vector_type(4))) unsigned int u32x4;
typedef __attribute__((ext_vector_type(8))) int i32x8;
typedef __attribute__((ext_vector_type(4))) int i32x4;

#define BSZ 64
#define TLEN 40000

// phase-1 chunking: steps t=1..29999 in chunks of 32 (last chunk = 15 steps)
#define NC1 938
#define LAST1_LEN 15
// superchunks over phase-1 chunks: 29 full (32 chunks) + last (10 chunks)
#define NSUP 30
#define LAST_SUP 10
// phase-2 chunking: steps t=30000..39999 in chunks of 32 (last = 16 steps)
#define NC2 313

// workspace layout (float offsets)
#define WS_A      0          // A[16]
#define WS_GS     16         // k*bb
#define WS_INV    17         // 1/norm1 (written by pass2b)
#define WS_MST    64         // A^1..A^32          (32*16)
#define WS_MST2   576        // (A^32)^1..^32      (32*16)
#define WS_TLAST  1088       // A^303              (16)
#define WS_MU     1104       // mu^1..mu^32        (32*4) as (re0,im0,re1,im1)
#define WS_K      1232       // K_ext 128x68 row-major (8704 floats)
#define WS_V      9936       // v[938][64][4]
#define WS_Y1     250064     // Ystart1[938][64][4]
#define WS_Y2     490192     // Ystart2[313][64][4]
#define WS_VS     570320     // Vs[30][64][4]
#define WS_YSUP   578000     // Ysuper[30][64][4]
// total: 585680 floats = ~2.35 MB

// output layout (float offsets): y(B,T,2 cplx), b(T), a(T), ysum(B,T,1 cplx)
#define OUT_Y   0
#define OUT_B   10240000
#define OUT_A   10280000
#define OUT_YS  10320000

// ---------------------------------------------------------------------------

__device__ __forceinline__ void mm4(const float* A, const float* B, float* C) {
  for (int r = 0; r < 4; r++)
    for (int c = 0; c < 4; c++) {
      float s = 0.f;
      for (int q = 0; q < 4; q++) s += A[r * 4 + q] * B[q * 4 + c];
      C[r * 4 + c] = s;
    }
}

__device__ __forceinline__ float4 affine_step(const float* M, float4 y, float4 v) {
  float4 r;
  r.x = M[0]  * y.x + M[1]  * y.y + M[2]  * y.z + M[3]  * y.w + v.x;
  r.y = M[4]  * y.x + M[5]  * y.y + M[6]  * y.z + M[7]  * y.w + v.y;
  r.z = M[8]  * y.x + M[9]  * y.y + M[10] * y.z + M[11] * y.w + v.z;
  r.w = M[12] * y.x + M[13] * y.y + M[14] * y.z + M[15] * y.w + v.w;
  return r;
}

// ---------------------------------------------------------------------------
// Kernel 0: build A, its powers, K_ext, mu powers.  Single thread (one-time).
// ---------------------------------------------------------------------------
__global__ void k_setup(const float* tau_y, const float* omega,
                        const float* a0, const float* b0, float* ws) {
  if (threadIdx.x != 0 || blockIdx.x != 0) return;
  const float tau = tau_y[0];
  const float om0 = omega[0], om1 = omega[1];
  const float a0v = a0[0], b0v = b0[0];
  const float dt = 0.1f;
  const float k = dt / tau;
  const float bb = b0v / (1.f + b0v);
  const float den = 1.f + a0v;
  const float twopi = 6.28318530717958647692f;
  const float ci0 = twopi * tau * om0;
  const float ci1 = twopi * tau * om1;
  const float gs = k * bb;

  float A[16];
  A[0] = 1.f + k * (1.f / den - 1.f);      A[1] = -k * ci0 / den;
  A[2] = -k * bb;                          A[3] = 0.f;
  A[4] = k * ci0 / den;                    A[5] = 1.f + k * (1.f / den - 1.f + bb);
  A[6] = 0.f;                              A[7] = 0.f;
  A[8] = -k * bb;                          A[9] = 0.f;
  A[10] = 1.f + k * (1.f / den - 1.f);     A[11] = -k * ci1 / den;
  A[12] = 0.f;                             A[13] = 0.f;
  A[14] = k * ci1 / den;                   A[15] = 1.f + k * (1.f / den - 1.f + bb);

  for (int i = 0; i < 16; i++) ws[WS_A + i] = A[i];
  ws[WS_GS] = gs;

  // A^1..A^32
  float Apow[32][16];
  for (int i = 0; i < 16; i++) Apow[0][i] = A[i];
  for (int j = 1; j < 32; j++) mm4(A, Apow[j - 1], Apow[j]);
  for (int j = 0; j < 32; j++)
    for (int i = 0; i < 16; i++) ws[WS_MST + j * 16 + i] = Apow[j][i];

  // (A^32)^1..^32 and A^303 = A^15 * (A^32)^9
  float Mp[32][16];
  for (int i = 0; i < 16; i++) Mp[0][i] = Apow[31][i];
  for (int p = 1; p < 32; p++) mm4(Apow[31], Mp[p - 1], Mp[p]);
  for (int p = 0; p < 32; p++)
    for (int i = 0; i < 16; i++) ws[WS_MST2 + p * 16 + i] = Mp[p][i];
  float T29[16];
  mm4(Apow[14], Mp[8], T29);
  for (int i = 0; i < 16; i++) ws[WS_TLAST + i] = T29[i];

  // mu^1..mu^32 per channel, stored (re0,im0,re1,im1)
  const float mr0 = 1.f, mi0 = k * ci0, mr1 = 1.f, mi1 = k * ci1;
  float cr0 = mr0, cim0 = mi0, cr1 = mr1, cim1 = mi1;
  for (int j = 0; j < 32; j++) {
    ws[WS_MU + 4 * j + 0] = cr0;  ws[WS_MU + 4 * j + 1] = cim0;
    ws[WS_MU + 4 * j + 2] = cr1;  ws[WS_MU + 4 * j + 3] = cim1;
    float n0r = cr0 * mr0 - cim0 * mi0, n0i = cr0 * mi0 + cim0 * mr0;
    float n1r = cr1 * mr1 - cim1 * mi1, n1i = cr1 * mi1 + cim1 * mr1;
    cr0 = n0r; cim0 = n0i; cr1 = n1r; cim1 = n1i;
  }

  // K_ext: rows 4(j-1)+r; cols 0..63 = gs*A^{j-i}[:,{0,2}], cols 64..67 = A^j
  const float Id[16] = {1,0,0,0, 0,1,0,0, 0,0,1,0, 0,0,0,1};
  float* Kp = ws + WS_K;
  for (int i = 0; i < 128 * 68; i++) Kp[i] = 0.f;
  for (int j = 1; j <= 32; j++) {
    for (int r = 0; r < 4; r++) {
      const int row = 4 * (j - 1) + r;
      for (int i = 1; i <= j; i++) {
        const int p = j - i;
        const float* Bm = (p == 0) ? Id : &Apow[p - 1][0];
        Kp[row * 68 + 2 * (i - 1) + 0] = gs * Bm[r * 4 + 0];
        Kp[row * 68 + 2 * (i - 1) + 1] = gs * Bm[r * 4 + 2];
      }
      for (int q = 0; q < 4; q++)
        Kp[row * 68 + 64 + q] = Apow[j - 1][r * 4 + q];
    }
  }
}

// ---------------------------------------------------------------------------
// Kernel 1: per (chunk, batch) input accumulation v_c (y starts at 0).
// ---------------------------------------------------------------------------
__global__ void k_pass1(const float* __restrict__ X, float* ws) {
  const int id = blockIdx.x * blockDim.x + threadIdx.x;
  if (id >= NC1 * BSZ) return;
  const int c = id >> 6, b = id & 63;
  float A[16];
#pragma unroll
  for (int i = 0; i < 16; i++) A[i] = ws[WS_A + i];
  const float gs = ws[WS_GS];
  const int len = (c == NC1 - 1) ? LAST1_LEN : 32;
  const int t0 = c * 32;
  const float* Xp = X + ((size_t)b * TLEN + t0 + 1) * 2;
  __builtin_prefetch(Xp, 0, 0);
  float4 y = make_float4(0.f, 0.f, 0.f, 0.f);
  for (int i = 0; i < len; i++) {
    const float z0 = Xp[2 * i], z1 = Xp[2 * i + 1];
    float4 v = make_float4(gs * z0, 0.f, gs * z1, 0.f);
    y = affine_step(A, y, v);
  }
  *(float4*)(ws + WS_V + ((size_t)c * BSZ + b) * 4) = y;
}

// ---------------------------------------------------------------------------
// Kernel 2a: superchunk reduction: Vs[s] = fold of 32 chunk-steps from 0.
// ---------------------------------------------------------------------------
__global__ void k_pass2a(float* ws) {
  const int id = blockIdx.x * blockDim.x + threadIdx.x;
  if (id >= NSUP * BSZ) return;
  const int s = id >> 6, b = id & 63;
  float M32[16], A15[16];
#pragma unroll
  for (int i = 0; i < 16; i++) M32[i] = ws[WS_MST + 31 * 16 + i];
#pragma unroll
  for (int i = 0; i < 16; i++) A15[i] = ws[WS_MST + 14 * 16 + i];
  const int cBeg = s * 32;
  const int nc = (s == NSUP - 1) ? LAST_SUP : 32;
  float4 y = make_float4(0.f, 0.f, 0.f, 0.f);
  for (int i = 0; i < nc; i++) {
    const int c = cBeg + i;
    float4 vv = *(const float4*)(ws + WS_V + ((size_t)c * BSZ + b) * 4);
    if (c == NC1 - 1) y = affine_step(A15, y, vv);
    else              y = affine_step(M32, y, vv);
  }
  *(float4*)(ws + WS_VS + ((size_t)s * BSZ + b) * 4) = y;
}

// ---------------------------------------------------------------------------
// Kernel 2b: sequential combine over 30 superchunks (64 lanes = batches),
// then the diagonal phase-2 boundary chain and 1/norm1.
// ---------------------------------------------------------------------------
__global__ void k_pass2b(float* ws) {
  const int b = threadIdx.x;
  if (b >= BSZ) return;
  float M1024[16], T29[16];
#pragma unroll
  for (int i = 0; i < 16; i++) M1024[i] = ws[WS_MST2 + 31 * 16 + i];
#pragma unroll
  for (int i = 0; i < 16; i++) T29[i] = ws[WS_TLAST + i];
  float4 y = make_float4(0.f, 0.f, 0.f, 0.f);
  for (int s = 0; s < NSUP; s++) {
    *(float4*)(ws + WS_YSUP + ((size_t)s * BSZ + b) * 4) = y;
    float4 vv = *(const float4*)(ws + WS_VS + ((size_t)s * BSZ + b) * 4);
    if (s == NSUP - 1) y = affine_step(T29, y, vv);
    else               y = affine_step(M1024, y, vv);
  }
  const float4 y29999 = y;  // state at t = 29999
  // phase-2 chunk-start states: multiply by mu^32 per chunk
  const float m32r0 = ws[WS_MU + 31 * 4 + 0], m32i0 = ws[WS_MU + 31 * 4 + 1];
  const float m32r1 = ws[WS_MU + 31 * 4 + 2], m32i1 = ws[WS_MU + 31 * 4 + 3];
  for (int c2 = 0; c2 < NC2; c2++) {
    *(float4*)(ws + WS_Y2 + ((size_t)c2 * BSZ + b) * 4) = y;
    float n0r = y.x * m32r0 - y.y * m32i0, n0i = y.x * m32i0 + y.y * m32r0;
    float n1r = y.z * m32r1 - y.w * m32i1, n1i = y.z * m32i1 + y.w * m32r1;
    y = make_float4(n0r, n0i, n1r, n1i);
  }
  if (b == 0) {
    const float m1r = ws[WS_MU + 0], m1i = ws[WS_MU + 1];
    const float re = y29999.x * m1r - y29999.y * m1i;
    const float im = y29999.x * m1i + y29999.y * m1r;
    ws[WS_INV] = 1.0f / sqrtf(re * re + im * im);
  }
}

// ---------------------------------------------------------------------------
// Kernel 2c: replay superchunks -> per-chunk start states Ystart1[c][b].
// ---------------------------------------------------------------------------
__global__ void k_pass2c(float* ws) {
  const int id = blockIdx.x * blockDim.x + threadIdx.x;
  if (id >= NSUP * BSZ) return;
  const int s = id >> 6, b = id & 63;
  float M32[16];
#pragma unroll
  for (int i = 0; i < 16; i++) M32[i] = ws[WS_MST + 31 * 16 + i];
  const int cBeg = s * 32;
  const int nc = (s == NSUP - 1) ? LAST_SUP : 32;
  float4 y = *(const float4*)(ws + WS_YSUP + ((size_t)s * BSZ + b) * 4);
  for (int i = 0; i < nc; i++) {
    const int c = cBeg + i;
    *(float4*)(ws + WS_Y1 + ((size_t)c * BSZ + b) * 4) = y;
    if (i + 1 < nc) {
      float4 vv = *(const float4*)(ws + WS_V + ((size_t)c * BSZ + b) * 4);
      y = affine_step(M32, y, vv);
    }
  }
}

// ---------------------------------------------------------------------------
// TDM descriptor helpers (D# per CDNA5 ISA).
// 6-arg builtin form (clang-23 / therock-10.0):
//   (u32x4 g0, i32x8 g1, i32x4 g2, i32x4 g3, i32x8 g4, i32 cpol)
// Groups 2/3 (and the trailing 8-dword operand) are only used for >=3D
// tensors / gather mode; zero-filled here.
// ---------------------------------------------------------------------------
__device__ __forceinline__ void tdm_load_1d(unsigned int lds_byte,
                                            unsigned long long gaddr,
                                            unsigned int ndw) {
  u32x4 g0;
  g0[0] = 1u;                                            // count=1 (user D#)
  g0[1] = lds_byte;                                      // lds_addr (bytes)
  g0[2] = (unsigned int)(gaddr & 0xFFFFFFFFull);         // global_addr lo
  g0[3] = (unsigned int)((gaddr >> 32) & 0x1FFFFFFull)   // global_addr hi
          | (2u << 30);                                  // type = 2 (image)
  i32x8 g1;
  g1[0] = (int)(2u << 16);                               // data_size = 4B
  g1[1] = (int)((ndw & 0xFFFFu) << 16);                  // tensor_dim0 lo16
  g1[2] = (int)(ndw >> 16);                              // tensor_dim0 hi16 | tensor_dim1=0
  g1[3] = (int)((ndw & 0xFFFFu) << 16);                  // tile_dim0 = ndw
  g1[4] = 0;                                             // tile_dim1/2 unused
  g1[5] = (int)ndw;                                      // dim0 stride (unused for 1D)
  g1[6] = 0;
  g1[7] = 0;
  i32x4 g2 = {0, 0, 0, 0};
  i32x4 g3 = {0, 0, 0, 0};
  i32x8 g4 = {0, 0, 0, 0, 0, 0, 0, 0};
  __builtin_amdgcn_tensor_load_to_lds(g0, g1, g2, g3, g4, 0);
}

__device__ __forceinline__ void tdm_load_2d_pad(unsigned int lds_byte,
                                                unsigned long long gaddr,
                                                unsigned int dim0_dw,   // contiguous dwords/row
                                                unsigned int rows,
                                                unsigned int row_stride_dw,
                                                unsigned int pad_interval_code,
                                                unsigned int pad_amount_code) {
  u32x4 g0;
  g0[0] = 1u;
  g0[1] = lds_byte;
  g0[2] = (unsigned int)(gaddr & 0xFFFFFFFFull);
  g0[3] = (unsigned int)((gaddr >> 32) & 0x1FFFFFFull) | (2u << 30);
  i32x8 g1;
  g1[0] = (int)((2u << 16)                     // data_size = 4B
                | (1u << 20)                   // pad_enable
                | (pad_interval_code << 22)    // pad interval
                | (pad_amount_code << 25));    // pad amount
  g1[1] = (int)((dim0_dw & 0xFFFFu) << 16);                    // tensor_dim0 lo16
  g1[2] = (int)((dim0_dw >> 16) | ((rows & 0xFFFFu) << 16));   // dim0 hi | tensor_dim1 lo
  g1[3] = (int)((rows >> 16) | ((dim0_dw & 0xFFFFu) << 16));   // dim1 hi | tile_dim0
  g1[4] = (int)(rows & 0xFFFFu);                               // tile_dim1 | tile_dim2=0
  g1[5] = (int)row_stride_dw;                                  // tensor_dim0_stride lo32
  g1[6] = 0;                                                   // stride0 hi | stride1 lo
  g1[7] = 0;
  i32x4 g2 = {0, 0, 0, 0};
  i32x4 g3 = {0, 0, 0, 0};
  i32x8 g4 = {0, 0, 0, 0, 0, 0, 0, 0};
  __builtin_amdgcn_tensor_load_to_lds(g0, g1, g2, g3, g4, 0);
}

// ---------------------------------------------------------------------------
// Kernel 3: phase-1 replay as WMMA GEMM.  One workgroup (8 waves) per chunk.
// S(128 x 64) = K_ext(128 x 68) * [U(64 x 64) ; Ystart(4 x 64)]
// K (8704 dw) and U (64x64 dw, LDS-padded to stride 68) staged via TDM;
// Ystart dropped into the 4-dword pad slots so one LDS layout serves all.
// ---------------------------------------------------------------------------
__global__ __launch_bounds__(256) void k_phase1_wmma(const float* __restrict__ X,
                                                     const float* __restrict__ ws,
                                                     float* __restrict__ out) {
  __shared__ float Kl[128 * 68];   // row-major, stride 68 (conflict-free)
  __shared__ float Ul[64 * 68];    // [batch][64 z dwords | 4 Ystart dwords]
  const int c = blockIdx.x;
  const int tid = threadIdx.x;
  const int len = (c == NC1 - 1) ? LAST1_LEN : 32;
  const int t0 = c * 32;

  // Wave 0: issue both TDM loads, then drain TENSORcnt before the barrier.
  if (tid < 32) {
    const unsigned int k_lds = (unsigned int)(unsigned long long)(&Kl[0]);
    const unsigned int u_lds = (unsigned int)(unsigned long long)(&Ul[0]);
    tdm_load_1d(k_lds, (unsigned long long)(ws + WS_K), 128u * 68u);
    // U: 64 dwords per batch (32 steps x 2 ch), 64 batches, row stride 80000
    // dwords; pad 4 dwords (code 3) after every 64 dwords (code 5) -> stride 68.
    tdm_load_2d_pad(u_lds,
                    (unsigned long long)(X + (size_t)(t0 + 1) * 2),
                    64u, 64u, (unsigned int)(TLEN * 2), 5u, 3u);
    __builtin_amdgcn_s_wait_tensorcnt(0);
  }
  // Chunk-start states into the pad slots: Ul[b*68 + 64..67].
  if (tid < BSZ) {
    float4 v = *(const float4*)(ws + WS_Y1 + ((size_t)c * BSZ + tid) * 4);
    *(float4*)(&Ul[tid * 68 + 64]) = v;
  }
  __syncthreads();

  const float inv = ws[WS_INV];
  const int lane = tid & 31;
  const int m = lane & 15, half = lane >> 4;
  const int rt = __builtin_amdgcn_readfirstlane(tid >> 5);  // wave-uniform row tile

  for (int ct = 0; ct < 4; ct++) {
    v8f acc = {};
    // kt = 16: A^j * y_start term (always present)
    {
      const int kc = 64 + 2 * half;
      v2f a, bm;
      a.x = Kl[(16 * rt + m) * 68 + kc];
      a.y = Kl[(16 * rt + m) * 68 + kc + 1];
      bm.x = Ul[(16 * ct + m) * 68 + kc];
      bm.y = Ul[(16 * ct + m) * 68 + kc + 1];
      acc = __builtin_amdgcn_wmma_f32_16x16x4_f32(false, a, false, bm,
                                                  (short)0, acc, false, false);
    }
    // triangular input tiles: kt = 0 .. 2*rt+1 (scalar bound -> scalar branch)
    const int ktmax = 2 * rt + 1;
    for (int kt = 0; kt <= ktmax; kt++) {
      const int kc = 4 * kt + 2 * half;
      v2f a, bm;
      a.x = Kl[(16 * rt + m) * 68 + kc];
      a.y = Kl[(16 * rt + m) * 68 + kc + 1];
      bm.x = Ul[(16 * ct + m) * 68 + kc];
      bm.y = Ul[(16 * ct + m) * 68 + kc + 1];
      acc = __builtin_amdgcn_wmma_f32_16x16x4_f32(false, a, false, bm,
                                                  (short)0, acc, false, false);
    }
    // D layout: VGPR v -> row 16*rt + v + 8*half, lane%16 -> column (batch)
    const int batch = 16 * ct + m;
    const size_t baseY = (size_t)batch * TLEN;
    const int jA = 4 * rt + 2 * half + 1;
    if (jA <= len) {
      const int t = t0 + jA;
      float4 o = make_float4(acc[0] * inv, acc[1] * inv, acc[2] * inv, acc[3] * inv);
      *(float4*)(out + OUT_Y + (baseY + t) * 4) = o;
      float2 s = make_float2((acc[0] + acc[2]) * inv, (acc[1] + acc[3]) * inv);
      *(float2*)(out + OUT_YS + (baseY + t + 1) * 2) = s;
    }
    const int jB = jA + 1;
    if (jB <= len) {
      const int t = t0 + jB;
      float4 o = make_float4(acc[4] * inv, acc[5] * inv, acc[6] * inv, acc[7] * inv);
      *(float4*)(out + OUT_Y + (baseY + t) * 4) = o;
      float2 s = make_float2((acc[4] + acc[6]) * inv, (acc[5] + acc[7]) * inv);
      *(float2*)(out + OUT_YS + (baseY + t + 1) * 2) = s;
    }
  }
}

// ---------------------------------------------------------------------------
// Kernel 4: phase 2, fully parallel elementwise: y_t = mu^{j} .* y_start.
// ---------------------------------------------------------------------------
__global__ void k_phase2(const float* __restrict__ ws, float* __restrict__ out) {
  const int id = blockIdx.x * blockDim.x + threadIdx.x;
  if (id >= BSZ * 10000) return;
  const int b = id / 10000;
  const int r = id - b * 10000;
  const int t = 30000 + r;
  const int c2 = r >> 5, jj = r & 31;
  float4 y  = *(const float4*)(ws + WS_Y2 + ((size_t)c2 * BSZ + b) * 4);
  float4 mu = *(const float4*)(ws + WS_MU + (size_t)jj * 4);
  const float inv = ws[WS_INV];
  const float yr0 = y.x * mu.x - y.y * mu.y, yi0 = y.x * mu.y + y.y * mu.x;
  const float yr1 = y.z * mu.z - y.w * mu.w, yi1 = y.z * mu.w + y.w * mu.z;
  *(float4*)(out + OUT_Y + ((size_t)b * TLEN + t) * 4) =
      make_float4(yr0 * inv, yi0 * inv, yr1 * inv, yi1 * inv);
  if (t + 1 < TLEN) {
    *(float2*)(out + OUT_YS + ((size_t)b * TLEN + t + 1) * 2) =
        make_float2((yr0 + yr1) * inv, (yi0 + yi1) * inv);
  }
}

// ---------------------------------------------------------------------------
// Kernel 5: b[t], a[t] step functions + t=0 / t=1 zero outputs.
// ---------------------------------------------------------------------------
__global__ void k_ba(const float* __restrict__ a0, const float* __restrict__ b0,
                     float* __restrict__ out) {
  const int id = blockIdx.x * blockDim.x + threadIdx.x;
  if (id < TLEN) {
    const float bv = (id < 30000) ? b0[0] : 0.0f;
    const float av = (id < 30000) ? a0[0] : 0.0f;
    out[OUT_B + id] = bv;
    out[OUT_A + id] = av;
  } else if (id < TLEN + BSZ) {
    const int b = id - TLEN;
    *(float4*)(out + OUT_Y + (size_t)b * TLEN * 4) = make_float4(0.f, 0.f, 0.f, 0.f);
    *(float2*)(out + OUT_YS + (size_t)b * TLEN * 2) = make_float2(0.f, 0.f);
    *(float2*)(out + OUT_YS + ((size_t)b * TLEN + 1) * 2) = make_float2(0.f, 0.f);
  }
}

// ---------------------------------------------------------------------------

extern "C" void kernel_launch(void* const* d_in, const int* in_sizes, int n_in,
                              void* d_out, int out_size, void* d_ws, size_t ws_size,
                              hipStream_t stream) {
  (void)in_sizes; (void)n_in; (void)out_size; (void)ws_size;
  const float* X     = (const float*)d_in[0];
  const float* tau_y = (const float*)d_in[1];
  const float* omega = (const float*)d_in[2];
  const float* a0    = (const float*)d_in[3];
  const float* b0    = (const float*)d_in[4];
  float* out = (float*)d_out;
  float* ws  = (float*)d_ws;

  k_setup<<<1, 1, 0, stream>>>(tau_y, omega, a0, b0, ws);
  k_pass1<<<(NC1 * BSZ + 255) / 256, 256, 0, stream>>>(X, ws);
  k_pass2a<<<(NSUP * BSZ + 255) / 256, 256, 0, stream>>>(ws);
  k_pass2b<<<1, 64, 0, stream>>>(ws);
  k_pass2c<<<(NSUP * BSZ + 255) / 256, 256, 0, stream>>>(ws);
  k_phase1_wmma<<<NC1, 256, 0, stream>>>(X, ws, out);
  k_phase2<<<(BSZ * 10000 + 255) / 256, 256, 0, stream>>>(ws, out);
  k_ba<<<(TLEN + BSZ + 255) / 256, 256, 0, stream>>>(a0, b0, out);
}